// PhysicsCalc_6588479832614
// MI455X (gfx1250) — compile-verified
//
#include <hip/hip_runtime.h>
#include <hip/hip_bf16.h>

typedef __attribute__((ext_vector_type(2))) float v2f;
typedef __attribute__((ext_vector_type(8))) float v8f;

#define B_ 512
#define S_ 8000
#define RAD 256          // truncated Gaussian radius (8.5 sigma, tail ~1e-17)
#define TAPS 513
#define KWIN 528         // 16 + 2*RAD, per-wave K extent (multiple of 4)
#define NWAVES 10        // waves per block
#define BLK_COLS 160     // NWAVES * 16 output cols per block
#define LDS_SPAN 672     // BLK_COLS + 2*RAD
#define LDS_STRIDE 676   // bank-skewed row stride (36-bank skew per row)

// ---- output layout (flat f32 concatenation, reference return order) ----
static constexpr long long OFF_DVOUT = 0;                          // 512*7700
static constexpr long long OFF_DVT   = 512LL * 7700;
static constexpr long long OFF_DV2T  = OFF_DVT  + 512LL * 8000;
static constexpr long long OFF_A     = OFF_DV2T + 512LL * 8000;
static constexpr long long OFF_K     = OFF_A + 512;
static constexpr long long OFF_E     = OFF_K + 512;
static constexpr long long OFF_G     = OFF_E + 512;
static constexpr long long OFF_T     = OFF_G + 512;
static constexpr long long OFF_DVL   = OFF_T   + 512LL * 8000;
static constexpr long long OFF_DVR   = OFF_DVL + 512LL * 8001;
static constexpr long long OFF_TOUT  = OFF_DVR + 512LL * 8001;

// ---------------------------------------------------------------------------
// Kernel 0: Gaussian weights. Sum over the full 7001 taps (matches reference
// normalization), then write the 513 central normalized taps.
// ---------------------------------------------------------------------------
__global__ void gauss_weights_kernel(float* __restrict__ wts) {
    __shared__ float red[256];
    const int t = threadIdx.x;
    float s = 0.f;
    for (int i = t; i < 7001; i += 256) {
        float x = (float)(i - 3500);
        s += expf(-x * x * (1.0f / 1800.0f));
    }
    red[t] = s;
    __syncthreads();
    for (int off = 128; off > 0; off >>= 1) {
        if (t < off) red[t] += red[t + off];
        __syncthreads();
    }
    const float denom = red[0];
    for (int i = t; i < TAPS; i += 256) {
        float x = (float)(i - RAD);
        wts[i] = expf(-x * x * (1.0f / 1800.0f)) / denom;
    }
}

// ---------------------------------------------------------------------------
// Kernel 1: per-row params. One wave32 per row of b (512x256).
// clip(relu(dot+bias), lo, hi) == clamp(dot+bias, lo, hi) since lo > 0.
// ---------------------------------------------------------------------------
__global__ void params_kernel(const float* __restrict__ bmat,
                              const float* __restrict__ kw, const float* __restrict__ kb,
                              const float* __restrict__ aw, const float* __restrict__ ab,
                              const float* __restrict__ ew, const float* __restrict__ eb,
                              const float* __restrict__ gw, const float* __restrict__ gb,
                              float* __restrict__ pk, float* __restrict__ pa,
                              float* __restrict__ pe, float* __restrict__ pg,
                              float* __restrict__ out_a, float* __restrict__ out_k,
                              float* __restrict__ out_e, float* __restrict__ out_g) {
    const int row  = blockIdx.x * 8 + (threadIdx.x >> 5);
    const int lane = threadIdx.x & 31;
    const float* br = bmat + row * 256;
    float sk = 0.f, sa = 0.f, se = 0.f, sg = 0.f;
    for (int i = lane; i < 256; i += 32) {
        const float bv = br[i];
        sk = fmaf(bv, kw[i], sk);
        sa = fmaf(bv, aw[i], sa);
        se = fmaf(bv, ew[i], se);
        sg = fmaf(bv, gw[i], sg);
    }
    for (int off = 16; off > 0; off >>= 1) {
        sk += __shfl_down(sk, off, 32);
        sa += __shfl_down(sa, off, 32);
        se += __shfl_down(se, off, 32);
        sg += __shfl_down(sg, off, 32);
    }
    if (lane == 0) {
        const float kv = fminf(fmaxf(sk + kb[0], 0.1f),   1.0f);
        const float av = fminf(fmaxf(sa + ab[0], 0.001f), 0.1f);
        const float ev = fminf(fmaxf(se + eb[0], 0.005f), 0.1f);
        const float gv = fminf(fmaxf(sg + gb[0], 0.5f),   2.0f);
        pk[row] = kv; pa[row] = av; pe[row] = ev; pg[row] = gv;
        out_k[row] = kv * 1000.0f;
        out_a[row] = av;
        out_e[row] = ev;
        out_g[row] = gv;
    }
}

// ---------------------------------------------------------------------------
// Kernel 2: Gaussian conv as Toeplitz GEMM on V_WMMA_F32_16X16X4_F32.
// Block: 320 threads = 10 waves; each wave owns one 16x16 output tile.
// Grid: (S/160, B/16) = (50, 32).
// Middle K-loop (kk in [16,508]) is guard-free: Toeplitz index t0 = kk+kofs-ncol
// is provably in [1,511], so B is a pair of unconditional ds loads.
// ---------------------------------------------------------------------------
__global__ __launch_bounds__(32 * NWAVES)
void conv_wmma_kernel(const float* __restrict__ v,
                      const float* __restrict__ wts,
                      float* __restrict__ vs) {
    __shared__ float lv[16 * LDS_STRIDE];
    __shared__ float lw[TAPS];

    const int m0    = blockIdx.y << 4;       // 16 rows per tile
    const int s0blk = blockIdx.x * BLK_COLS; // 160 output cols per block
    const int tid   = threadIdx.x;

    // Stage 16 rows x 672 cols of input (zero-padded at edges) into LDS.
    for (int idx = tid; idx < 16 * LDS_SPAN; idx += 32 * NWAVES) {
        const int row = idx / LDS_SPAN;
        const int col = idx - row * LDS_SPAN;
        const int g   = s0blk - RAD + col;
        float val = 0.f;
        if (g >= 0 && g < S_) val = v[(m0 + row) * S_ + g];
        lv[row * LDS_STRIDE + col] = val;
    }
    for (int idx = tid; idx < TAPS; idx += 32 * NWAVES) lw[idx] = wts[idx];
    __syncthreads();

    const int wave = tid >> 5;
    const int lane = tid & 31;
    const int mrow = lane & 15;
    const int kofs = (lane >> 4) << 1;   // lanes 16-31 hold K pair {2,3}
    const int ncol = lane & 15;

    const float* arow  = &lv[mrow * LDS_STRIDE + (wave << 4)];
    const float* wbase = lw + kofs - ncol;   // Toeplitz: B[t][n] = w[t-n]

    v8f c = {};

    // Left band edge (t0 may go negative): guarded.
#pragma unroll
    for (int kk = 0; kk < 16; kk += 4) {
        v2f a;
        a.x = arow[kk + kofs];
        a.y = arow[kk + kofs + 1];
        const int t0 = kk + kofs - ncol;
        v2f bb;
        bb.x = ((unsigned)t0       <= (unsigned)(TAPS - 1)) ? lw[t0]     : 0.f;
        bb.y = ((unsigned)(t0 + 1) <= (unsigned)(TAPS - 1)) ? lw[t0 + 1] : 0.f;
        c = __builtin_amdgcn_wmma_f32_16x16x4_f32(false, a, false, bb,
                                                  (short)0, c, false, false);
    }

    // Guard-free middle: t0 in [1,511] always.
#pragma unroll 4
    for (int kk = 16; kk <= 508; kk += 4) {
        v2f a = *(const v2f*)(arow + kk + kofs);   // aligned ds_load_b64
        v2f bb;
        bb.x = wbase[kk];
        bb.y = wbase[kk + 1];
        c = __builtin_amdgcn_wmma_f32_16x16x4_f32(false, a, false, bb,
                                                  (short)0, c, false, false);
    }

    // Right band edge (t0+1 may exceed 512): guarded.
#pragma unroll
    for (int kk = 512; kk < KWIN; kk += 4) {
        v2f a;
        a.x = arow[kk + kofs];
        a.y = arow[kk + kofs + 1];
        const int t0 = kk + kofs - ncol;
        v2f bb;
        bb.x = ((unsigned)t0       <= (unsigned)(TAPS - 1)) ? lw[t0]     : 0.f;
        bb.y = ((unsigned)(t0 + 1) <= (unsigned)(TAPS - 1)) ? lw[t0 + 1] : 0.f;
        c = __builtin_amdgcn_wmma_f32_16x16x4_f32(false, a, false, bb,
                                                  (short)0, c, false, false);
    }

    // C/D layout: VGPR r -> (M=r, N=lane) lanes 0-15, (M=8+r, N=lane-16) lanes 16-31
    const int colg  = s0blk + (wave << 4) + ncol;
    const int rbase = m0 + ((lane >> 4) << 3);
#pragma unroll
    for (int r = 0; r < 8; ++r)
        vs[(rbase + r) * S_ + colg] = c[r];
}

// ---------------------------------------------------------------------------
// Kernel 3: pointwise physics + all output streams (memory-bound pass).
// ---------------------------------------------------------------------------
__global__ __launch_bounds__(256)
void pointwise_kernel(const float* __restrict__ vs,
                      const float* __restrict__ v0,
                      const float* __restrict__ pk, const float* __restrict__ pa,
                      const float* __restrict__ pe, const float* __restrict__ pg,
                      const int* __restrict__ use_cond,
                      float* __restrict__ out) {
    const int gid = blockIdx.x * blockDim.x + threadIdx.x;
    if (gid >= B_ * S_) return;
    const int row = gid / S_;
    const int s   = gid - row * S_;

    const float vsc = vs[gid];
    const float vm  = (s > 0)      ? vs[gid - 1] : 0.f;   // dv_l[:-1]
    const float vp  = (s < S_ - 1) ? vs[gid + 1] : 0.f;   // dv_r[1:]

    const float dt  = 1.6f / 7999.0f;
    const float dvt = (vp - vm) / (2.0f * dt);
    const float dv2 = (vp + vm - 2.0f * vsc) / (dt * dt);

    const float k = pk[row], a = pa[row], e = pe[row], g = pg[row];

    const float vn  = (vsc == 0.f) ? 0.001f : vsc;
    float inv = 1.0f / vn;
    if (inv != inv) inv = 0.f;  // NaN -> 0

    const float tl1 = vsc * k * dvt * (1.0f - 2.0f * vsc + a);
    const float tl2 = inv * dvt * dvt;
    const float tlt = 1000.0f * tl1 + tl2 - dv2;
    const float tr1 = vsc * (vsc - g * k * 1000.0f * (1.0f - vsc) * (vsc - a));
    const float trt = e * (tr1 + g * dvt);
    float dvo = tlt - trt;
    if (use_cond[0] != 0 && v0[gid] <= 0.f) dvo = 0.f;

    out[OFF_DVT  + gid] = dvt;
    out[OFF_DV2T + gid] = dv2;
    out[OFF_T    + gid] = 1.6f;
    out[OFF_TOUT + gid] = 1.6f;
    out[OFF_DVL + (long long)row * 8001 + s + 1] = vsc;   // left-pad
    out[OFF_DVR + (long long)row * 8001 + s]     = vsc;   // right-pad
    if (s == 0) {
        out[OFF_DVL + (long long)row * 8001]        = 0.f;
        out[OFF_DVR + (long long)row * 8001 + 8000] = 0.f;
    }

    int oi = -1;
    if (s >= 100 && s < 950)        oi = s - 100;
    else if (s >= 1050 && s < 7900) oi = 850 + (s - 1050);
    if (oi >= 0)
        out[OFF_DVOUT + (long long)row * 7700 + oi] = logf(dvo * dvo + 1e-5f);
}

// ---------------------------------------------------------------------------
extern "C" void kernel_launch(void* const* d_in, const int* in_sizes, int n_in,
                              void* d_out, int out_size, void* d_ws, size_t ws_size,
                              hipStream_t stream) {
    const float* v_out = (const float*)d_in[0];
    const float* bmat  = (const float*)d_in[1];
    const float* k_w   = (const float*)d_in[2];
    const float* k_b   = (const float*)d_in[3];
    const float* a_w   = (const float*)d_in[4];
    const float* a_b   = (const float*)d_in[5];
    const float* e_w   = (const float*)d_in[6];
    const float* e_b   = (const float*)d_in[7];
    const float* g_w   = (const float*)d_in[8];
    const float* g_b   = (const float*)d_in[9];
    const int*   ucond = (const int*)d_in[10];

    float* out = (float*)d_out;
    float* ws  = (float*)d_ws;

    float* vs  = ws;                        // 512*8000 f32 smoothed signal
    float* wts = ws + (size_t)B_ * S_;      // 513 taps (+pad)
    float* pk  = wts + 1024;
    float* pa  = pk + 512;
    float* pe  = pa + 512;
    float* pg  = pe + 512;

    gauss_weights_kernel<<<1, 256, 0, stream>>>(wts);

    params_kernel<<<64, 256, 0, stream>>>(bmat, k_w, k_b, a_w, a_b, e_w, e_b,
                                          g_w, g_b, pk, pa, pe, pg,
                                          out + OFF_A, out + OFF_K,
                                          out + OFF_E, out + OFF_G);

    conv_wmma_kernel<<<dim3(S_ / BLK_COLS, B_ / 16), 32 * NWAVES, 0, stream>>>(
        v_out, wts, vs);

    pointwise_kernel<<<(B_ * S_ + 255) / 256, 256, 0, stream>>>(
        vs, v_out, pk, pa, pe, pg, ucond, out);
}